// ACTIONVLAD_44933947851351
// MI455X (gfx1250) — compile-verified
//
#include <hip/hip_runtime.h>
#include <math.h>

// ---------------------------------------------------------------------------
// ActionVLAD for MI455X (gfx1250, wave32).
// Matmuls use V_WMMA_F32_16X16X4_F32 (full f32 precision, 2048 FLOP/instr).
// x1 (48MB) is read exactly once; Wt (128KB) stays L2-resident; per-frame
// vlad accumulators live in registers (128 VGPRs/thread, CDNA5 allows 1024).
// Pipeline: double-buffered LDS x-tiles with register staging + prefetch of
// tile it+2 (global_prefetch_b8), softmax parallelized over all 256 threads.
// Workspace layout (floats): ws1[30*32768] raw vlad, ws2[30*64] scales,
// wt[512*64] transposed conv weights  => ~3.9 MB total.
// ---------------------------------------------------------------------------

#define T_FRAMES 30
#define CDIM     512
#define NK       64
#define NPIX     784
#define NT       16          // pixels per tile
#define NITER    49          // 49*16 == 784, no tail
#define EPSN     1e-12f
#define XSTR     20          // LDS stride for x tile  (16B-aligned float4 rows)
#define ASTR     18          // LDS stride for a tile  (8B-aligned v2f reads)
#define LSTR     66          // LDS stride for logit partials

typedef float v2f __attribute__((ext_vector_type(2)));
typedef float v8f __attribute__((ext_vector_type(8)));

// ---- kernel 0: Wt[c][k] = conv_w[k][c]  (coalesced writes) -----------------
__global__ __launch_bounds__(256) void k_transposeW(const float* __restrict__ w,
                                                    float* __restrict__ wt) {
    int idx = blockIdx.x * 256 + threadIdx.x;      // idx = c*64 + k
    if (idx < CDIM * NK) {
        int c = idx >> 6, k = idx & 63;
        wt[idx] = w[k * CDIM + c];
    }
}

// ---- x-tile register staging helpers (8 x float4 per thread = 32 KB/tile) --
__device__ __forceinline__ void load_tile(const float* __restrict__ xfrm, int n0,
                                          int tid, float4* stg) {
#pragma unroll
    for (int s = 0; s < 8; ++s) {
        int idx = tid + s * 256;
        int c = idx >> 2, q = idx & 3;
        stg[s] = *reinterpret_cast<const float4*>(xfrm + c * NPIX + n0 + q * 4);
    }
}
__device__ __forceinline__ void store_tile(float* xb, int tid, const float4* stg) {
#pragma unroll
    for (int s = 0; s < 8; ++s) {
        int idx = tid + s * 256;
        int c = idx >> 2, q = idx & 3;
        *reinterpret_cast<float4*>(&xb[c * XSTR + q * 4]) = stg[s];
    }
}

// ---- kernel 1: per-frame VLAD core -----------------------------------------
__global__ __launch_bounds__(256) void k_vlad_main(const float* __restrict__ x1,
                                                   const float* __restrict__ cent,
                                                   const float* __restrict__ wt,
                                                   const float* __restrict__ bias,
                                                   float* __restrict__ ws1) {
    __shared__ __align__(16) float xbuf[2][CDIM * XSTR]; // normalized x tiles [c][p]
    __shared__ float logitP[2 * 16 * LSTR];              // [chalf][p][k]
    __shared__ float a_s[NK * ASTR];                     // softmax assignments [k][p]
    __shared__ float redn[16 * 17];                      // per-pixel sumsq partials
    __shared__ float invn[16];                           // per-pixel 1/||x||
    __shared__ float pmax[16 * 17];                      // softmax max partials
    __shared__ float psum[16 * 17];                      // softmax sum partials
    __shared__ float smax[16];
    __shared__ float sinvS[16];
    __shared__ float asum_s[NK];
    __shared__ float bias_s[NK];

    const int tid  = threadIdx.x;
    const int lane = tid & 31;
    const int w    = tid >> 5;            // wave id 0..7
    const int l16  = lane & 15;
    const int lh   = lane >> 4;           // lane half (WMMA K/M interleave)
    const int t    = blockIdx.x;

    const int m  = w & 3;                 // vlad k-row tile / logits k-col tile
    const int h  = w >> 2;                // vlad c-half   / logits channel-half

    const int sp = tid >> 4;              // softmax: pixel handled by this thread
    const int sq = tid & 15;              // softmax: 16-thread group index

    if (tid < NK) bias_s[tid] = bias[tid];

    // vlad accumulators: 16 tiles of 16x16 f32 per wave (k-rows m*16.., c-half h)
    v8f vacc[16];
#pragma unroll
    for (int ct = 0; ct < 16; ++ct)
#pragma unroll
        for (int j = 0; j < 8; ++j) vacc[ct][j] = 0.f;

    float asum_r = 0.f;                   // per-thread (tid<64) sum_n a[k,n]

    const float* xfrm = x1 + (size_t)t * CDIM * NPIX;

    // ---- pipeline prologue: stage tile 0 ----
    float4 stg[8];
    load_tile(xfrm, 0, tid, stg);
    store_tile(&xbuf[0][0], tid, stg);
    __syncthreads();

    for (int it = 0; it < NITER; ++it) {
        const int cur = it & 1;
        float* xb = &xbuf[cur][0];

        // issue next tile's global loads now; latency hides behind all compute
        if (it + 1 < NITER) load_tile(xfrm, (it + 1) * NT, tid, stg);
        // warm WGP$/L2 for the tile after that (global_prefetch_b8)
        if (it + 2 < NITER) {
            const float* pf = xfrm + (it + 2) * NT;
            for (int c = tid; c < CDIM; c += 256)
                __builtin_prefetch(pf + c * NPIX, 0, 3);
        }

        // ---- per-pixel L2 norm over 512 channels ----
        {
            float s = 0.f;
            for (int c = sq * 32; c < sq * 32 + 32; ++c) {
                float v = xb[c * XSTR + sp];
                s += v * v;
            }
            redn[sp * 17 + sq] = s;
        }
        __syncthreads();
        if (tid < 16) {
            float s = 0.f;
            for (int part = 0; part < 16; ++part) s += redn[tid * 17 + part];
            invn[tid] = 1.f / fmaxf(sqrtf(s), EPSN);
        }
        __syncthreads();
        for (int idx = tid; idx < CDIM * NT; idx += 256) {
            int c = idx >> 4, p = idx & 15;
            xb[c * XSTR + p] *= invn[p];
        }
        __syncthreads();

        // ---- logits^T tile: D[p][k] = sum_c xn[c][p] * Wt[c][k] ----
        // 8 waves = 4 k-tiles (m) x 2 channel-halves (h); partials summed later.
        {
            v8f lacc = {0.f, 0.f, 0.f, 0.f, 0.f, 0.f, 0.f, 0.f};
            const int cb = h * 256;
            for (int cc = 0; cc < 256; cc += 4) {
                const int c = cb + cc + 2 * lh;     // ISA A-layout: K {0,2}/{1,3}
                v2f a, b;
                a.x = xb[c * XSTR + l16];
                a.y = xb[(c + 1) * XSTR + l16];
                b.x = wt[c * NK + m * 16 + l16];    // lane-contiguous global read
                b.y = wt[(c + 1) * NK + m * 16 + l16];
                lacc = __builtin_amdgcn_wmma_f32_16x16x4_f32(
                    false, a, false, b, (short)0, lacc, false, false);
            }
#pragma unroll
            for (int j = 0; j < 8; ++j) {           // D row interleave by 8
                int p = j + 8 * lh;
                logitP[(h * 16 + p) * LSTR + m * 16 + l16] = lacc[j];
            }
        }
        __syncthreads();

        // ---- softmax over 64 clusters, all 256 threads (16/pixel x 4 k) ----
        float lv[4];
        {
            float mx4 = -3.4e38f;
#pragma unroll
            for (int i = 0; i < 4; ++i) {
                int k = sq * 4 + i;
                lv[i] = logitP[sp * LSTR + k] + logitP[(16 + sp) * LSTR + k] + bias_s[k];
                mx4 = fmaxf(mx4, lv[i]);
            }
            pmax[sp * 17 + sq] = mx4;
        }
        __syncthreads();
        if (tid < 16) {
            float mxx = -3.4e38f;
            for (int qq = 0; qq < 16; ++qq) mxx = fmaxf(mxx, pmax[tid * 17 + qq]);
            smax[tid] = mxx;
        }
        __syncthreads();
        {
            float mm = smax[sp];
            float s4 = 0.f;
#pragma unroll
            for (int i = 0; i < 4; ++i) {
                float e = __expf(lv[i] - mm);
                a_s[(sq * 4 + i) * ASTR + sp] = e;
                s4 += e;
            }
            psum[sp * 17 + sq] = s4;
        }
        __syncthreads();
        if (tid < 16) {
            float s = 0.f;
            for (int qq = 0; qq < 16; ++qq) s += psum[tid * 17 + qq];
            sinvS[tid] = 1.f / s;
        }
        __syncthreads();
        {
            float si = sinvS[sp];
#pragma unroll
            for (int i = 0; i < 4; ++i) a_s[(sq * 4 + i) * ASTR + sp] *= si;
        }
        __syncthreads();

        if (tid < NK) {                   // accumulate sum_n a[k,n]
            float s = 0.f;
            for (int p = 0; p < NT; ++p) s += a_s[tid * ASTR + p];
            asum_r += s;
        }

        // ---- vlad accumulate: D[k][c] += sum_p a[k][p] * xn[c][p] ----
#pragma unroll
        for (int kk = 0; kk < 16; kk += 4) {
            const int col = kk + 2 * lh;
            v2f a;
            a.x = a_s[(m * 16 + l16) * ASTR + col];
            a.y = a_s[(m * 16 + l16) * ASTR + col + 1];
#pragma unroll
            for (int ct = 0; ct < 16; ++ct) {
                const int c0 = h * 256 + ct * 16;
                v2f b = *reinterpret_cast<v2f*>(&xb[(c0 + l16) * XSTR + col]); // 8B aligned
                vacc[ct] = __builtin_amdgcn_wmma_f32_16x16x4_f32(
                    false, a, false, b, (short)0, vacc[ct], false, false);
            }
        }

        // ---- drain staged registers into the other LDS buffer ----
        if (it + 1 < NITER) store_tile(&xbuf[1 - cur][0], tid, stg);
        __syncthreads();
    }

    if (tid < NK) asum_s[tid] = asum_r;
    __syncthreads();

    // ---- subtract a_sum * centroid, store raw vlad (coalesced 64B runs) ----
    float* outT = ws1 + (size_t)t * (NK * CDIM);
#pragma unroll
    for (int ct = 0; ct < 16; ++ct) {
#pragma unroll
        for (int j = 0; j < 8; ++j) {
            int k = m * 16 + j + 8 * lh;
            int c = h * 256 + ct * 16 + l16;
            outT[k * CDIM + c] = vacc[ct][j] - asum_s[k] * cent[k * CDIM + c];
        }
    }
}

// ---- kernel 2: per-frame intra + global L2 scale factors -------------------
__global__ __launch_bounds__(256) void k_vlad_norm(const float* __restrict__ ws1,
                                                   float* __restrict__ ws2) {
    __shared__ float red[NK * 4];
    __shared__ float sInv[NK];
    __shared__ float sSS[NK];
    __shared__ float gInv;
    const int tid = threadIdx.x, t = blockIdx.x;
    const float* v = ws1 + (size_t)t * NK * CDIM;
    int k = tid >> 2, part = tid & 3;
    float s = 0.f;
    for (int c = part * 128; c < part * 128 + 128; ++c) {
        float x = v[k * CDIM + c];
        s += x * x;
    }
    red[tid] = s;                          // tid == k*4 + part
    __syncthreads();
    if (tid < NK) {
        float ss = red[tid * 4] + red[tid * 4 + 1] + red[tid * 4 + 2] + red[tid * 4 + 3];
        sSS[tid]  = ss;
        sInv[tid] = 1.f / fmaxf(sqrtf(ss), EPSN);
    }
    __syncthreads();
    if (tid == 0) {
        float gs = 0.f;
        for (int kk = 0; kk < NK; ++kk) gs += sSS[kk] * sInv[kk] * sInv[kk];
        gInv = 1.f / fmaxf(sqrtf(gs), EPSN);
    }
    __syncthreads();
    if (tid < NK) ws2[t * NK + tid] = sInv[tid] * gInv;
}

// ---- kernel 3: deterministic sum over frames -------------------------------
__global__ __launch_bounds__(256) void k_vlad_sum(const float* __restrict__ ws1,
                                                  const float* __restrict__ ws2,
                                                  float* __restrict__ out) {
    int i = blockIdx.x * 256 + threadIdx.x;
    if (i < NK * CDIM) {
        int k = i >> 9;
        float s = 0.f;
        for (int t = 0; t < T_FRAMES; ++t)
            s += ws1[(size_t)t * NK * CDIM + i] * ws2[t * NK + k];
        out[i] = s;
    }
}

extern "C" void kernel_launch(void* const* d_in, const int* in_sizes, int n_in,
                              void* d_out, int out_size, void* d_ws, size_t ws_size,
                              hipStream_t stream) {
    const float* x1    = (const float*)d_in[0];   // [30,512,28,28]
    const float* cent  = (const float*)d_in[1];   // [64,512]
    const float* convw = (const float*)d_in[2];   // [64,512]
    const float* convb = (const float*)d_in[3];   // [64]

    float* ws  = (float*)d_ws;
    float* ws1 = ws;                              // 30*32768 raw vlad
    float* ws2 = ws1 + T_FRAMES * NK * CDIM;      // 30*64 scales
    float* wt  = ws2 + T_FRAMES * NK;             // 512*64 transposed weights

    k_transposeW<<<128, 256, 0, stream>>>(convw, wt);
    k_vlad_main<<<T_FRAMES, 256, 0, stream>>>(x1, cent, wt, convb, ws1);
    k_vlad_norm<<<T_FRAMES, 256, 0, stream>>>(ws1, ws2);
    k_vlad_sum<<<128, 256, 0, stream>>>(ws1, ws2, (float*)d_out);
}